// FirstStageNetwork_88837103550989
// MI455X (gfx1250) — compile-verified
//
#include <hip/hip_runtime.h>

typedef __attribute__((ext_vector_type(2))) float v2f;
typedef __attribute__((ext_vector_type(8))) float v8f;

#define N_NODES 50000
#define IN_DIM  128
#define OUT_DIM 256
#define N_EDGES 800000

// ---------------------------------------------------------------------------
// Degree: self-loop contributes 1 to every node.
// ---------------------------------------------------------------------------
__global__ void k_deg_init(unsigned int* __restrict__ deg) {
    int i = blockIdx.x * 256 + threadIdx.x;
    if (i < N_NODES) deg[i] = 1u;
}

__global__ void k_deg_count(const int* __restrict__ ei, unsigned int* __restrict__ deg) {
    int e = blockIdx.x * 256 + threadIdx.x;
    if (e < N_EDGES) {
        int dst = ei[2 * e + 1];
        atomicAdd(&deg[dst], 1u);   // global_atomic_add_u32
    }
}

__global__ void k_dinv(const unsigned int* __restrict__ deg, float* __restrict__ dinv) {
    int i = blockIdx.x * 256 + threadIdx.x;
    if (i < N_NODES) dinv[i] = rsqrtf((float)deg[i]);   // deg >= 1 always
}

// ---------------------------------------------------------------------------
// hs = (x @ W^T) * dinv[row]  via V_WMMA_F32_16X16X4_F32 (full fp32 accuracy).
// Also seeds acc (= d_out) with hs: the self-loop contribution.
// One wave per 16x16 output tile; 3125 x 16 tiles = 50000 tiles, 8 waves/block.
//
// A (16x4 f32): lanes 0-15 -> M=lane, VGPR{0,1}=K{0,1}; lanes 16-31 -> K{2,3}.
// B (4x16 f32): lanes 0-15 -> N=lane, VGPR{0,1}=K{0,1}; lanes 16-31 -> K{2,3}.
// C/D (16x16 f32): VGPR r, lanes 0-15 -> M=r,  N=lane;
//                          lanes 16-31 -> M=r+8, N=lane-16.
// ---------------------------------------------------------------------------
__global__ __launch_bounds__(256) void k_gemm_wmma(
        const float* __restrict__ x, const float* __restrict__ W,
        const float* __restrict__ dinv,
        float* __restrict__ hs, float* __restrict__ acc) {
    const int wave   = threadIdx.x >> 5;
    const int lane   = threadIdx.x & 31;
    const int tile   = blockIdx.x * 8 + wave;   // 0..49999
    const int tile_m = tile >> 4;               // 0..3124
    const int tile_n = tile & 15;               // 0..15
    const int m0     = tile_m * 16;
    const int n0     = tile_n * 16;
    const int half   = lane >> 4;               // 0: K pair {0,1}, 1: {2,3}
    const int l16    = lane & 15;

    const float* arow = x + (size_t)(m0 + l16) * IN_DIM + 2 * half;
    const float* brow = W + (size_t)(n0 + l16) * IN_DIM + 2 * half;

    v8f c = {};
#pragma unroll
    for (int k = 0; k < IN_DIM; k += 4) {
        v2f a = *(const v2f*)(arow + k);
        v2f b = *(const v2f*)(brow + k);
        // 8 args: (neg_a, A, neg_b, B, c_mod, C, reuse_a, reuse_b)
        c = __builtin_amdgcn_wmma_f32_16x16x4_f32(false, a, false, b,
                                                  (short)0, c, false, false);
    }

#pragma unroll
    for (int r = 0; r < 8; ++r) {
        int row = m0 + r + 8 * half;
        int col = n0 + l16;
        float v = c[r] * dinv[row];
        size_t o = (size_t)row * OUT_DIM + col;
        hs[o]  = v;   // scaled features, gathered by the edge kernel
        acc[o] = v;   // accumulator seeded with self-loop term
    }
}

// ---------------------------------------------------------------------------
// Per edge: acc[dst] += hs[src].  ONE WAVE PER EDGE (32 lanes x 8 floats).
// Edge id is made wave-uniform via readfirstlane so the src/dst pair lowers
// to a single scalar s_load_b64 per edge (SMEM/KMcnt path) instead of 32
// redundant vector loads -> saves ~400 MB of index traffic on the
// bottleneck kernel. Gather is two coalesced global_load_b128 per lane
// (wave reads the full 1KB row; each lane reads a contiguous 32B slice); scatter is
// hardware global_atomic_add_f32 (unsafeAtomicAdd, no CAS loop).
// hs (51 MB) and acc (51 MB) are L2-resident (192 MB) -> L2-bandwidth bound.
// ---------------------------------------------------------------------------
__global__ __launch_bounds__(256) void k_edge_scatter(
        const int* __restrict__ ei,
        const float* __restrict__ hs, float* __restrict__ acc) {
    const int wave = __builtin_amdgcn_readfirstlane(threadIdx.x >> 5); // uniform
    const int lane = threadIdx.x & 31;
    const int e    = blockIdx.x * 8 + wave;     // one edge per wave, uniform
    // e < N_EDGES by exact grid sizing (100000 blocks * 8 waves)
    const int src = ei[2 * e + 0];              // scalar load (uniform address)
    const int dst = ei[2 * e + 1];

    const float* s = hs  + (size_t)src * OUT_DIM + lane * 8;
    float*       p = acc + (size_t)dst * OUT_DIM + lane * 8;
    const float4 v0 = *(const float4*)(s + 0);
    const float4 v1 = *(const float4*)(s + 4);
    unsafeAtomicAdd(p + 0, v0.x);
    unsafeAtomicAdd(p + 1, v0.y);
    unsafeAtomicAdd(p + 2, v0.z);
    unsafeAtomicAdd(p + 3, v0.w);
    unsafeAtomicAdd(p + 4, v1.x);
    unsafeAtomicAdd(p + 5, v1.y);
    unsafeAtomicAdd(p + 6, v1.z);
    unsafeAtomicAdd(p + 7, v1.w);
}

// ---------------------------------------------------------------------------
// out = relu(acc * dinv[row] + b), float4-vectorized, in place on d_out.
// ---------------------------------------------------------------------------
__global__ void k_finalize(float* __restrict__ out, const float* __restrict__ dinv,
                           const float* __restrict__ b) {
    int tid = blockIdx.x * 256 + threadIdx.x;       // N_NODES*64 threads
    if (tid >= N_NODES * 64) return;
    int node = tid >> 6;
    int c4   = (tid & 63) * 4;
    float*  p  = out + (size_t)node * OUT_DIM + c4;
    float4  v  = *(float4*)p;
    float4  bb = *(const float4*)(b + c4);
    float   d  = dinv[node];
    v.x = fmaxf(fmaf(v.x, d, bb.x), 0.0f);
    v.y = fmaxf(fmaf(v.y, d, bb.y), 0.0f);
    v.z = fmaxf(fmaf(v.z, d, bb.z), 0.0f);
    v.w = fmaxf(fmaf(v.w, d, bb.w), 0.0f);
    *(float4*)p = v;
}

// ---------------------------------------------------------------------------
extern "C" void kernel_launch(void* const* d_in, const int* in_sizes, int n_in,
                              void* d_out, int out_size, void* d_ws, size_t ws_size,
                              hipStream_t stream) {
    const float* x  = (const float*)d_in[0];    // (50000, 128) f32
    const int*   ei = (const int*)d_in[1];      // (800000, 2) int
    const float* W  = (const float*)d_in[2];    // (256, 128) f32
    const float* b  = (const float*)d_in[3];    // (256,) f32
    float* out = (float*)d_out;                 // (50000, 256) f32

    // Workspace layout: hs[50000*256] f32 | deg[50000] u32 | dinv[50000] f32
    float*        hs   = (float*)d_ws;
    unsigned int* deg  = (unsigned int*)(hs + (size_t)N_NODES * OUT_DIM);
    float*        dinv = (float*)(deg + N_NODES);

    k_deg_init <<<(N_NODES + 255) / 256, 256, 0, stream>>>(deg);
    k_deg_count<<<(N_EDGES + 255) / 256, 256, 0, stream>>>(ei, deg);
    k_dinv     <<<(N_NODES + 255) / 256, 256, 0, stream>>>(deg, dinv);

    // 50000 tiles / 8 waves per 256-thread block = 6250 blocks (exact)
    k_gemm_wmma<<<6250, 256, 0, stream>>>(x, W, dinv, hs, out);

    // 800000 edges / 8 waves per block = 100000 blocks (exact)
    k_edge_scatter<<<100000, 256, 0, stream>>>(ei, hs, out);

    k_finalize<<<(N_NODES * 64 + 255) / 256, 256, 0, stream>>>(out, dinv, b);
}